// ClusterLoss_927712936124
// MI455X (gfx1250) — compile-verified
//
#include <hip/hip_runtime.h>
#include <hip/hip_bf16.h>
#include <math.h>

// Problem constants (match reference)
#define BB 2048
#define DD 512
#define CC 20
#define LAM 250.0f            // D / (B * EPS) = 512 / (2048 * 1e-3)
#define NT 528                // upper-tri 16x16 tiles in 32x32 tile grid
#define TPB 8                 // tiles (waves) per block
#define NBLK (NT / TPB)       // 66 blocks per gram job

typedef __attribute__((ext_vector_type(2))) float v2f;
typedef __attribute__((ext_vector_type(8))) float v8f;

// ---------------- ws layout ----------------
// [0      , 32)      : float acc[8] (0: head Σlog diag, 1: gtcr Σlog diag, 2: ce rowsum, 3: reg sum)
// [256    , 8448)    : int   labels[B]
// [16384  , ... )    : float grams[(C+1) * D * D]   (~22 MB)
#define WS_ACC_OFF     0
#define WS_LABELS_OFF  256
#define WS_GRAMS_OFF   16384

__global__ void init_acc_kernel(float* acc) {
    if (threadIdx.x < 8) acc[threadIdx.x] = 0.0f;
}

// argmax(yhat, axis=1) -> labels  (first-max tie rule like jnp.argmax)
__global__ void labels_kernel(const float* __restrict__ yhat, int* __restrict__ labels) {
    int row = blockIdx.x * blockDim.x + threadIdx.x;
    if (row >= BB) return;
    const float* y = yhat + (size_t)row * CC;
    float best = y[0];
    int bi = 0;
    for (int j = 1; j < CC; ++j) {
        float v = y[j];
        if (v > best) { best = v; bi = j; }
    }
    labels[row] = bi;
}

// decode upper-triangular tile index -> (i, j), i <= j, 32x32 tile grid
__device__ __forceinline__ void decode_tile(int t, int& i, int& j) {
    i = 0;
    while (true) {
        int rl = 32 - i;
        if (t < rl) break;
        t -= rl;
        ++i;
    }
    j = i + t;
}

// Core 16x16 Gram tile: K=B with 4 independent WMMA accumulator chains.
// MASKED is compile-time so the hot loop is branch-free either way.
template <bool MASKED>
__device__ __forceinline__ void gram_tile(const float* __restrict__ h,
                                          const float* __restrict__ maskf,
                                          float* __restrict__ G,
                                          int i, int j, float scale) {
    const int lane = threadIdx.x & 31;
    const int half = lane >> 4;        // 0|1
    const int m = lane & 15;
    const int colA = i * 16 + m;
    const int colB = j * 16 + m;

    v8f acc0 = {}, acc1 = {}, acc2 = {}, acc3 = {};
#pragma unroll 1
    for (int kb = 0; kb < BB; kb += 16) {
        v8f* accs[4] = {&acc0, &acc1, &acc2, &acc3};
#pragma unroll
        for (int u = 0; u < 4; ++u) {
            const int r0 = kb + 4 * u + 2 * half;
            float a0 = h[(size_t)r0 * DD + colA];
            float a1 = h[(size_t)(r0 + 1) * DD + colA];
            float b0 = h[(size_t)r0 * DD + colB];
            float b1 = h[(size_t)(r0 + 1) * DD + colB];
            if (MASKED) {              // 0/1 float mask from LDS (ds_load_b64 + pk_mul)
                a0 *= maskf[r0];
                a1 *= maskf[r0 + 1];
            }
            v2f a, b;
            a.x = a0; a.y = a1;
            b.x = b0; b.y = b1;
            *accs[u] = __builtin_amdgcn_wmma_f32_16x16x4_f32(
                false, a, false, b, (short)0, *accs[u], false, false);
        }
    }
    const v8f acc = (acc0 + acc1) + (acc2 + acc3);

#pragma unroll
    for (int r = 0; r < 8; ++r) {
        const int M = r + 8 * half;    // C/D layout: reg r, lanes16-31 offset by 8
        const int row = i * 16 + M;
        const int col = j * 16 + m;
        const float v = scale * acc[r] + ((row == col) ? 1.0f : 0.0f);
        G[(size_t)row * DD + col] = v;
        if (i != j) G[(size_t)col * DD + row] = v;   // symmetric mirror tile
    }
}

// Per-class masked grams: G_c = I + 0.5 * S_c. Grid = CC * NBLK, 256 threads (8 waves).
// One LDS mask per block, shared by the 8 tiles.
__global__ void gram_class_kernel(const float* __restrict__ h,
                                  const int* __restrict__ labels,
                                  float* __restrict__ grams) {
    const int job = blockIdx.x / NBLK;
    const int tbase = (blockIdx.x % NBLK) * TPB;

    __shared__ float maskf[BB];
    for (int t = threadIdx.x; t < BB; t += blockDim.x)
        maskf[t] = (labels[t] == job) ? 1.0f : 0.0f;
    __syncthreads();

    const int wave = threadIdx.x >> 5;
    int i, j;
    decode_tile(tbase + wave, i, j);
    gram_tile<true>(h, maskf, grams + (size_t)job * DD * DD, i, j, 0.5f);
}

// GTCR gram: M = I + lam * (h^T h). Grid = NBLK, 256 threads. No mask.
__global__ void gram_gtcr_kernel(const float* __restrict__ h,
                                 float* __restrict__ grams) {
    const int tbase = (blockIdx.x % NBLK) * TPB;
    const int wave = threadIdx.x >> 5;
    int i, j;
    decode_tile(tbase + wave, i, j);
    gram_tile<false>(h, nullptr, grams + (size_t)CC * DD * DD, i, j, LAM);
}

// One workgroup per SPD matrix: in-place right-looking Cholesky (lower triangle),
// accumulate sum(log(diag(L))) == 0.5 * logdet.
__global__ void cholesky_logdet_kernel(float* __restrict__ grams, float* __restrict__ acc) {
    const int job = blockIdx.x;
    float* A = grams + (size_t)job * DD * DD;

    __shared__ float colk[DD];
    __shared__ float s_inv;
    const int tid = threadIdx.x;
    const int nth = blockDim.x;

    float logsum = 0.0f;   // only meaningful on tid 0

    for (int k = 0; k < DD; ++k) {
        if (tid == 0) {
            float d = sqrtf(A[(size_t)k * DD + k]);
            logsum += logf(d);
            s_inv = 1.0f / d;
        }
        __syncthreads();
        const float inv = s_inv;
        for (int j = k + 1 + tid; j < DD; j += nth) {
            float v = A[(size_t)j * DD + k] * inv;
            A[(size_t)j * DD + k] = v;
            colk[j] = v;
        }
        __syncthreads();
        for (int j = k + 1 + tid; j < DD; j += nth) {
            const float cj = colk[j];
            float* rowp = A + (size_t)j * DD;
            for (int t = k + 1; t <= j; ++t) rowp[t] -= cj * colk[t];
        }
        __syncthreads();
    }

    if (tid == 0) {
        if (job < CC) atomicAdd(&acc[0], logsum);   // head: 0.5*logdet = Σ log diag
        else          atomicAdd(&acc[1], logsum);   // gtcr
    }
}

// cross entropy with soft targets: accumulate Σ_rows Σ_j softmax(yg)*log_softmax(yhat)
__global__ void ce_kernel(const float* __restrict__ yhat, const float* __restrict__ yg,
                          float* __restrict__ acc) {
    int row = blockIdx.x * blockDim.x + threadIdx.x;
    if (row >= BB) return;
    const float* p = yhat + (size_t)row * CC;
    const float* g = yg + (size_t)row * CC;

    float m1 = p[0], m2 = g[0];
    for (int j = 1; j < CC; ++j) { m1 = fmaxf(m1, p[j]); m2 = fmaxf(m2, g[j]); }
    float s1 = 0.0f, s2 = 0.0f;
    for (int j = 0; j < CC; ++j) { s1 += __expf(p[j] - m1); s2 += __expf(g[j] - m2); }
    const float lse1 = logf(s1);
    const float inv2 = 1.0f / s2;
    float dot = 0.0f;
    for (int j = 0; j < CC; ++j) {
        float tgt = __expf(g[j] - m2) * inv2;
        dot += tgt * (p[j] - m1 - lse1);
    }
    atomicAdd(&acc[2], dot);
}

// RegLoss: Σ over B*D of 0.5 - 0.5*erf(-(x+0.5)/(sqrt(2)*0.5))
__global__ void reg_kernel(const float* __restrict__ u, float* __restrict__ acc) {
    const int n = BB * DD;
    float part = 0.0f;
    for (int idx = blockIdx.x * blockDim.x + threadIdx.x; idx < n;
         idx += gridDim.x * blockDim.x) {
        float x = u[idx];
        float arg = -(x + 0.5f) * 1.41421356237309515f;  // /(sqrt(2)*0.5)
        part += 0.5f - 0.5f * erff(arg);
    }
    // wave32 reduction then one atomic per wave
    for (int off = 16; off > 0; off >>= 1)
        part += __shfl_down(part, off, 32);
    if ((threadIdx.x & 31) == 0) atomicAdd(&acc[3], part);
}

__global__ void finalize_kernel(const float* __restrict__ acc,
                                const float* __restrict__ lmbd,
                                float* __restrict__ out) {
    if (threadIdx.x == 0 && blockIdx.x == 0) {
        const float head = acc[0];
        const float gtcr = acc[1];
        const float ce = -acc[2] / (float)BB;
        const float reg = acc[3] / (float)(BB * DD);
        out[0] = head - gtcr + ce + lmbd[0] * reg;
    }
}

extern "C" void kernel_launch(void* const* d_in, const int* in_sizes, int n_in,
                              void* d_out, int out_size, void* d_ws, size_t ws_size,
                              hipStream_t stream) {
    const float* h    = (const float*)d_in[0];
    const float* yhat = (const float*)d_in[1];
    const float* yg   = (const float*)d_in[2];
    const float* u_zg = (const float*)d_in[3];
    const float* lmbd = (const float*)d_in[4];
    float* out = (float*)d_out;

    char* ws = (char*)d_ws;
    float* acc    = (float*)(ws + WS_ACC_OFF);
    int*   labels = (int*)(ws + WS_LABELS_OFF);
    float* grams  = (float*)(ws + WS_GRAMS_OFF);

    init_acc_kernel<<<1, 8, 0, stream>>>(acc);
    labels_kernel<<<(BB + 255) / 256, 256, 0, stream>>>(yhat, labels);
    gram_class_kernel<<<CC * NBLK, TPB * 32, 0, stream>>>(h, labels, grams);
    gram_gtcr_kernel<<<NBLK, TPB * 32, 0, stream>>>(h, grams);
    cholesky_logdet_kernel<<<CC + 1, 256, 0, stream>>>(grams, acc);
    ce_kernel<<<(BB + 255) / 256, 256, 0, stream>>>(yhat, yg, acc);
    reg_kernel<<<1024, 256, 0, stream>>>(u_zg, acc);
    finalize_kernel<<<1, 32, 0, stream>>>(acc, lmbd, out);
}